// Entmax_77378130805225
// MI455X (gfx1250) — compile-verified
//
#include <hip/hip_runtime.h>

// entmax-1.5 (reference-faithful sparsemax-tau + ^1.5), one workgroup per row.
// [2048 x 32000] f32. Memory-bound: ~524MB traffic -> ~22.5us floor @ 23.3TB/s.
// Sort-free: 4-way section search for the sparsemax threshold + exact tail fixup.
// CDNA5 path: global_load_async_to_lds_b128 staging + s_wait_asynccnt, wave32
// shuffle reductions, B128 LDS reads and B128 global stores.

#define NCOL     32000
#define TPB      1024
#define NGRP     8             // float4 groups per thread: 8*1024*4 = 32768 >= 32000
#define NWAVES   32
#define ACT_CAP  256
#define NCHUNK   4
#define CHUNK_F4 2048          // float4 per 32KB staging chunk
#define NF4      (NCOL / 4)    // 8000 float4 packets per row (exact)
#define NIT      5             // 4-way search iters: width 4^-5 = 2^-10

__device__ __forceinline__ void block_reduce4(float v0, float v1, float v2, float v3,
                                              float (*sred)[4], float* sbc,
                                              int lane, int wave,
                                              float& o0, float& o1, float& o2, float& o3) {
  // v0..v2 sum-reduced, v3 max-reduced across the 1024-thread block (wave32).
  #pragma unroll
  for (int off = 16; off > 0; off >>= 1) {
    v0 += __shfl_xor(v0, off, 32);
    v1 += __shfl_xor(v1, off, 32);
    v2 += __shfl_xor(v2, off, 32);
    v3  = fmaxf(v3, __shfl_xor(v3, off, 32));
  }
  if (lane == 0) { sred[wave][0] = v0; sred[wave][1] = v1; sred[wave][2] = v2; sred[wave][3] = v3; }
  __syncthreads();
  if (wave == 0) {
    float w0 = sred[lane][0], w1 = sred[lane][1], w2 = sred[lane][2], w3 = sred[lane][3];
    #pragma unroll
    for (int off = 16; off > 0; off >>= 1) {
      w0 += __shfl_xor(w0, off, 32);
      w1 += __shfl_xor(w1, off, 32);
      w2 += __shfl_xor(w2, off, 32);
      w3  = fmaxf(w3, __shfl_xor(w3, off, 32));
    }
    if (lane == 0) { sbc[0] = w0; sbc[1] = w1; sbc[2] = w2; sbc[3] = w3; }
  }
  __syncthreads();
  o0 = sbc[0]; o1 = sbc[1]; o2 = sbc[2]; o3 = sbc[3];
}

__global__ __launch_bounds__(TPB, 1)
void entmax15_kernel(const float* __restrict__ z, float* __restrict__ out) {
  __shared__ float4 sbuf[CHUNK_F4];      // async-copy staging (32KB)
  __shared__ float  sred[NWAVES][4];
  __shared__ float  sbc[4];
  __shared__ float  sact[ACT_CAP];
  __shared__ int    snact;
  __shared__ float  stau;

  const int tid  = threadIdx.x;
  const int lane = tid & 31;
  const int wave = tid >> 5;
  const size_t row = blockIdx.x;
  const float* Z = z + row * (size_t)NCOL;
  float* O = out + row * (size_t)NCOL;

  if (tid == 0) snact = 0;

  // ---- Stage row HBM -> LDS (async copy engine) -> VGPRs, 4 x 32KB chunks ----
  // Thread owns float4 packets e = tid + g*1024, g = 0..7; packet e is valid iff e < 8000.
  float4 r[NGRP];
  const unsigned lbase = (unsigned)(uintptr_t)(&sbuf[0]);
  const unsigned long long gbase = (unsigned long long)(uintptr_t)Z;
  const float NEG_INF = -__builtin_inff();
  for (int c = 0; c < NCHUNK; ++c) {
    #pragma unroll
    for (int q = 0; q < 2; ++q) {
      int e = (c * 2 + q) * 1024 + tid;           // global B128 packet index
      if (e < NF4) {
        unsigned lofs = lbase + (unsigned)((q * 1024 + tid) * 16);
        unsigned long long ga = gbase + (unsigned long long)e * 16ull;
        asm volatile("global_load_async_to_lds_b128 %0, %1, off"
                     :: "v"(lofs), "v"(ga) : "memory");
      }
    }
    asm volatile("s_wait_asynccnt 0" ::: "memory");
    __syncthreads();
    #pragma unroll
    for (int q = 0; q < 2; ++q) {
      int g = c * 2 + q;
      int e = g * 1024 + tid;
      if (e < NF4) {
        r[g] = sbuf[q * 1024 + tid];              // ds_load_b128, conflict-free
      } else {
        r[g] = make_float4(NEG_INF, NEG_INF, NEG_INF, NEG_INF);
      }
    }
    __syncthreads();   // buffer reused by next chunk
  }

  // ---- Row max M; threshold tau* with sum(relu(z - tau*)) = 1 lies in [M-1, M) ----
  float m = NEG_INF;
  #pragma unroll
  for (int g = 0; g < NGRP; ++g) {
    m = fmaxf(m, fmaxf(fmaxf(r[g].x, r[g].y), fmaxf(r[g].z, r[g].w)));
  }
  float d0, d1, d2, M;
  block_reduce4(0.f, 0.f, 0.f, m, sred, sbc, lane, wave, d0, d1, d2, M);

  // ---- 4-way section search: invariant f(lo) >= 1 > f(hi) ----
  float lo = M - 1.0f, hi = M;
  for (int it = 0; it < NIT; ++it) {
    float stp = 0.25f * (hi - lo);
    float p1 = lo + stp, p2 = lo + 2.0f * stp, p3 = lo + 3.0f * stp;
    float s1 = 0.f, s2 = 0.f, s3 = 0.f;
    #pragma unroll
    for (int g = 0; g < NGRP; ++g) {
      const float v[4] = { r[g].x, r[g].y, r[g].z, r[g].w };
      #pragma unroll
      for (int k = 0; k < 4; ++k) {
        s1 += fmaxf(v[k] - p1, 0.f);
        s2 += fmaxf(v[k] - p2, 0.f);
        s3 += fmaxf(v[k] - p3, 0.f);
      }
    }
    float f1, f2, f3, dm;
    block_reduce4(s1, s2, s3, 0.f, sred, sbc, lane, wave, f1, f2, f3, dm);
    if      (f3 >= 1.0f) { lo = p3; }
    else if (f2 >= 1.0f) { lo = p2; hi = p3; }
    else if (f1 >= 1.0f) { lo = p1; hi = p2; }
    else                 { hi = p1; }
  }

  // ---- Exact stats: K = #{z > hi}, S = sum{z > hi}, max{z <= lo}, actives in (lo, hi] ----
  float cnt = 0.f, sum = 0.f, mlo = NEG_INF;
  #pragma unroll
  for (int g = 0; g < NGRP; ++g) {
    const float v[4] = { r[g].x, r[g].y, r[g].z, r[g].w };
    #pragma unroll
    for (int k = 0; k < 4; ++k) {
      float x = v[k];
      if (x > hi)      { cnt += 1.0f; sum += x; }
      else if (x > lo) { int p = atomicAdd(&snact, 1); if (p < ACT_CAP) sact[p] = x; }
      else             { mlo = fmaxf(mlo, x); }
    }
  }
  float fcnt, fsum, dmy, gmlo;
  block_reduce4(cnt, sum, 0.f, mlo, sred, sbc, lane, wave, fcnt, fsum, dmy, gmlo);

  // ---- Exact tail fixup on thread 0 (reference off-by-one gather replicated) ----
  if (tid == 0) {
    int   K = (int)(fcnt + 0.5f);
    float S = fsum;
    int   nact = snact; if (nact > ACT_CAP) nact = ACT_CAP;
    // insertion sort descending (list is expected to be ~0-2 elements)
    for (int i = 1; i < nact; ++i) {
      float key = sact[i];
      int j = i - 1;
      while (j >= 0 && sact[j] < key) { sact[j + 1] = sact[j]; --j; }
      sact[j + 1] = key;
    }
    // support condition 1 + t*z_(t) > cs_t is prefix-true; top K certainly in support
    int   kmax = K;
    float cs = S;          // sum of current top-kmax
    for (int j = 0; j < nact; ++j) {
      float v = sact[j];
      float cs_t = cs + v;
      float t = (float)(K + j + 1);
      if (1.0f + t * v > cs_t) { kmax = K + j + 1; cs = cs_t; }
      else break;
    }
    // reference gathers cumsum at 0-based index kmax = sum of top (kmax+1), clamped at N
    int idx = kmax - K;
    float tau;
    if (idx >= nact && gmlo == NEG_INF) {
      tau = (cs - 1.0f) / (float)kmax;                       // kmax == N clamp path
    } else {
      float nextv = (idx < nact) ? sact[idx] : gmlo;         // (kmax+1)-th largest
      tau = (cs + nextv - 1.0f) / (float)kmax;
    }
    stau = tau;
  }
  __syncthreads();
  const float tau = stau;

  // ---- out = relu(z - tau)^1.5 = x * sqrt(x), B128 stores ----
  float4* O4 = (float4*)O;
  #pragma unroll
  for (int g = 0; g < NGRP; ++g) {
    int e = g * 1024 + tid;
    if (e < NF4) {
      float4 o;
      float x0 = fmaxf(r[g].x - tau, 0.f);
      float x1 = fmaxf(r[g].y - tau, 0.f);
      float x2 = fmaxf(r[g].z - tau, 0.f);
      float x3 = fmaxf(r[g].w - tau, 0.f);
      o.x = x0 * __builtin_sqrtf(x0);
      o.y = x1 * __builtin_sqrtf(x1);
      o.z = x2 * __builtin_sqrtf(x2);
      o.w = x3 * __builtin_sqrtf(x3);
      O4[e] = o;
    }
  }
}

extern "C" void kernel_launch(void* const* d_in, const int* in_sizes, int n_in,
                              void* d_out, int out_size, void* d_ws, size_t ws_size,
                              hipStream_t stream) {
  (void)n_in; (void)d_ws; (void)ws_size; (void)out_size;
  const float* z = (const float*)d_in[0];
  float* out = (float*)d_out;
  const int rows = in_sizes[0] / NCOL;   // 2048
  entmax15_kernel<<<dim3(rows), dim3(TPB), 0, stream>>>(z, out);
}